// UnitarySpectralFilter_8993661518152
// MI455X (gfx1250) — compile-verified
//
#include <hip/hip_runtime.h>

typedef float v2f __attribute__((ext_vector_type(2)));
typedef float v4f __attribute__((ext_vector_type(4)));
typedef float v8f __attribute__((ext_vector_type(8)));

#define USF_EPS 1e-8f

// ---------------------------------------------------------------------------
// Setup: compute the 4 complex circulant coefficients c_d from alpha.
//   c_d = 1/4 * sum_m exp(i*phi_m) * i^{(m*d) mod 4}
//   phi_m = alpha * atan(log(|k_m| + eps)),  k = fftfreq(4) = [0,.25,-.5,-.25]
// ws[0..3] = Re(c_0..c_3), ws[4..7] = Im(c_0..c_3)
// ---------------------------------------------------------------------------
__global__ void usf_coef_kernel(const float* __restrict__ alphap,
                                float* __restrict__ ws) {
  if (threadIdx.x != 0 || blockIdx.x != 0) return;
  const float alpha = alphap[0];
  const float km[4] = {USF_EPS, 0.25f + USF_EPS, 0.5f + USF_EPS, 0.25f + USF_EPS};
  float fr[4], fi[4];
#pragma unroll
  for (int m = 0; m < 4; ++m) {
    float phi = alpha * atanf(logf(km[m]));
    fr[m] = cosf(phi);
    fi[m] = sinf(phi);
  }
#pragma unroll
  for (int d = 0; d < 4; ++d) {
    float re = 0.f, im = 0.f;
#pragma unroll
    for (int m = 0; m < 4; ++m) {
      const int t = (m * d) & 3;   // multiply by i^t exactly
      if      (t == 0) { re += fr[m]; im += fi[m]; }
      else if (t == 1) { re -= fi[m]; im += fr[m]; }
      else if (t == 2) { re -= fr[m]; im -= fi[m]; }
      else             { re += fi[m]; im -= fr[m]; }
    }
    ws[d]     = 0.25f * re;
    ws[4 + d] = 0.25f * im;
  }
}

// ---------------------------------------------------------------------------
// Main: one V_WMMA_F32_16X16X4_F32 transforms 16 input rows.
//   A(16x4): row 2j+0 = Re(c_{(j-k)&3}), row 2j+1 = Im(c_{(j-k)&3}); rows 8-15 = 0
//   B(4x16): column n = input row n (4 floats). Per the ISA fragment layout each
//            lane loads one aligned 8B chunk; the wave covers a contiguous 256B.
//   D(16x16): lane n, v0..v7 = [re y0, im y0, re y1, im y1, ...] for row n
//             -> interleaved complex64 memory order, 2x NT B128 stores.
// 4 tiles per wave iteration: 4 independent NT b64 loads clause together
// before the first wait -> 1KB of loads in flight per wave.
// ---------------------------------------------------------------------------
__global__ void __launch_bounds__(256)
usf_wmma_kernel(const float* __restrict__ psi,
                const float* __restrict__ coef,
                float* __restrict__ out,
                long long num_tiles) {
  const int lane  = threadIdx.x & 31;
  const int m     = lane & 15;   // row index within tile / B column
  const int khalf = lane >> 4;   // 0 -> K=0,1   1 -> K=2,3

  // Build the A fragment once (hoisted out of the streaming loop).
  v2f a; a.x = 0.f; a.y = 0.f;
  if (m < 8) {
    const int j  = m >> 1;        // complex output index
    const int p  = m & 1;         // 0 = real row, 1 = imag row
    const int k0 = khalf << 1;    // first K this lane holds
    a.x = coef[p * 4 + ((j - k0)     & 3)];
    a.y = coef[p * 4 + ((j - k0 - 1) & 3)];
  }

  const long long wavesPerBlock = (long long)(blockDim.x >> 5);
  const long long waveId = (long long)blockIdx.x * wavesPerBlock + (threadIdx.x >> 5);
  const long long nWaves = (long long)gridDim.x * wavesPerBlock;

  const v8f czero = {0.f, 0.f, 0.f, 0.f, 0.f, 0.f, 0.f, 0.f};
  const long long laneLoadOff  = (long long)((m << 2) + (khalf << 1)); // floats
  const long long tilesAligned = num_tiles & ~3LL;

  // Main loop: 4 consecutive tiles (1KB) per wave iteration.
  for (long long t4 = waveId << 2; t4 < tilesAligned; t4 += (nWaves << 2)) {
    const float* base = psi + (t4 << 6) + laneLoadOff;
    v2f b0 = __builtin_nontemporal_load((const v2f*)(base));
    v2f b1 = __builtin_nontemporal_load((const v2f*)(base + 64));
    v2f b2 = __builtin_nontemporal_load((const v2f*)(base + 128));
    v2f b3 = __builtin_nontemporal_load((const v2f*)(base + 192));

    v8f d0 = __builtin_amdgcn_wmma_f32_16x16x4_f32(false, a, false, b0, (short)0, czero, false, false);
    v8f d1 = __builtin_amdgcn_wmma_f32_16x16x4_f32(false, a, false, b1, (short)0, czero, false, false);
    v8f d2 = __builtin_amdgcn_wmma_f32_16x16x4_f32(false, a, false, b2, (short)0, czero, false, false);
    v8f d3 = __builtin_amdgcn_wmma_f32_16x16x4_f32(false, a, false, b3, (short)0, czero, false, false);

    if (khalf == 0) {
      float* dst = out + (((t4 << 4) + (long long)m) << 3);
      v4f lo, hi;
      lo.x = d0[0]; lo.y = d0[1]; lo.z = d0[2]; lo.w = d0[3];
      hi.x = d0[4]; hi.y = d0[5]; hi.z = d0[6]; hi.w = d0[7];
      __builtin_nontemporal_store(lo, (v4f*)(dst));
      __builtin_nontemporal_store(hi, (v4f*)(dst + 4));
      lo.x = d1[0]; lo.y = d1[1]; lo.z = d1[2]; lo.w = d1[3];
      hi.x = d1[4]; hi.y = d1[5]; hi.z = d1[6]; hi.w = d1[7];
      __builtin_nontemporal_store(lo, (v4f*)(dst + 128));
      __builtin_nontemporal_store(hi, (v4f*)(dst + 132));
      lo.x = d2[0]; lo.y = d2[1]; lo.z = d2[2]; lo.w = d2[3];
      hi.x = d2[4]; hi.y = d2[5]; hi.z = d2[6]; hi.w = d2[7];
      __builtin_nontemporal_store(lo, (v4f*)(dst + 256));
      __builtin_nontemporal_store(hi, (v4f*)(dst + 260));
      lo.x = d3[0]; lo.y = d3[1]; lo.z = d3[2]; lo.w = d3[3];
      hi.x = d3[4]; hi.y = d3[5]; hi.z = d3[6]; hi.w = d3[7];
      __builtin_nontemporal_store(lo, (v4f*)(dst + 384));
      __builtin_nontemporal_store(hi, (v4f*)(dst + 388));
    }
  }

  // Tile remainder (< 4 tiles): one tile per wave, uniform per-wave predicate.
  {
    const long long t = tilesAligned + waveId;
    if (t < num_tiles) {
      const v2f* src = (const v2f*)(psi + (t << 6) + laneLoadOff);
      v2f b = __builtin_nontemporal_load(src);
      v8f d = __builtin_amdgcn_wmma_f32_16x16x4_f32(false, a, false, b, (short)0, czero, false, false);
      if (khalf == 0) {
        float* dst = out + (((t << 4) + (long long)m) << 3);
        v4f lo, hi;
        lo.x = d[0]; lo.y = d[1]; lo.z = d[2]; lo.w = d[3];
        hi.x = d[4]; hi.y = d[5]; hi.z = d[6]; hi.w = d[7];
        __builtin_nontemporal_store(lo, (v4f*)(dst));
        __builtin_nontemporal_store(hi, (v4f*)(dst + 4));
      }
    }
  }
}

// Scalar tail for row counts not divisible by 16 (not hit for 4096x4096).
__global__ void usf_tail_kernel(const float* __restrict__ psi,
                                const float* __restrict__ coef,
                                float* __restrict__ out,
                                long long rowStart, long long rowEnd) {
  long long r = rowStart + (long long)blockIdx.x * blockDim.x + threadIdx.x;
  if (r >= rowEnd) return;
  float x[4];
#pragma unroll
  for (int k = 0; k < 4; ++k) x[k] = psi[r * 4 + k];
#pragma unroll
  for (int j = 0; j < 4; ++j) {
    float re = 0.f, im = 0.f;
#pragma unroll
    for (int k = 0; k < 4; ++k) {
      const int d = (j - k) & 3;
      re += coef[d]     * x[k];
      im += coef[4 + d] * x[k];
    }
    out[r * 8 + 2 * j]     = re;
    out[r * 8 + 2 * j + 1] = im;
  }
}

extern "C" void kernel_launch(void* const* d_in, const int* in_sizes, int n_in,
                              void* d_out, int out_size, void* d_ws, size_t ws_size,
                              hipStream_t stream) {
  const float* psi   = (const float*)d_in[0];
  const float* alpha = (const float*)d_in[1];
  float* out = (float*)d_out;
  float* ws  = (float*)d_ws;

  const long long nElems = (long long)in_sizes[0];
  const long long nRows  = nElems / 4;        // 16,777,216
  const long long nTiles = nRows / 16;        // 1,048,576

  usf_coef_kernel<<<1, 32, 0, stream>>>(alpha, ws);

  if (nTiles > 0) {
    const int threads = 256;                  // 8 waves / block
    long long groups = (nTiles + 31) / 32;    // 4 tiles per wave per iter, 8 waves
    int blocks = (groups > 4096) ? 4096 : (int)groups;
    if (blocks < 1) blocks = 1;
    usf_wmma_kernel<<<blocks, threads, 0, stream>>>(psi, ws, out, nTiles);
  }

  const long long rowStart = nTiles * 16;
  if (rowStart < nRows) {
    const long long tailRows = nRows - rowStart;
    const int tb = 64;
    const int tg = (int)((tailRows + tb - 1) / tb);
    usf_tail_kernel<<<tg, tb, 0, stream>>>(psi, ws, out, rowStart, nRows);
  }
}